// TensorSquare_15461882266098
// MI455X (gfx1250) — compile-verified
//
#include <hip/hip_runtime.h>
#include <math.h>

typedef float v2f __attribute__((ext_vector_type(2)));
typedef float v8f __attribute__((ext_vector_type(8)));

#define M0   128
#define M1   64
#define NP0  8128   // triu(128,1) pairs
#define NP1  2016   // triu(64,1) pairs
#define XW   320    // input row width
#define OW   320    // output row width
#define X0S  132    // LDS stride, x0 tile   (132 % 64 == 4 -> conflict-free)
#define X1S  196    // LDS stride, x1 tile   (196 % 64 == 4)
#define FBS  130    // LDS stride, feature buffer (even -> b64 reads; %64==2)
#define R2   32     // rows per block, out0 kernel
#define R3   16     // rows per block, out1 kernel
#define KC   128    // max feature chunk

__device__ __forceinline__ v8f wmma4(v2f a, v2f b, v8f c) {
  // V_WMMA_F32_16X16X4_F32: D(16x16 f32) = A(16x4) * B(4x16) + C
  return __builtin_amdgcn_wmma_f32_16x16x4_f32(false, a, false, b, (short)0, c,
                                               false, false);
}

// ---------------------------------------------------------------------------
// Kernel 0: build packed (u | v<<8) triu-pair LUTs in workspace.
// ---------------------------------------------------------------------------
__global__ void build_luts(unsigned short* __restrict__ lut0,
                           unsigned short* __restrict__ lut1) {
  int k = blockIdx.x * blockDim.x + threadIdx.x;
  if (k < NP0) {
    int u = 0, rem = k;
    while (rem >= (M0 - 1 - u)) { rem -= (M0 - 1 - u); ++u; }
    lut0[k] = (unsigned short)(u | ((u + 1 + rem) << 8));
  }
  int k1 = k - NP0;
  if (k1 >= 0 && k1 < NP1) {
    int u = 0, rem = k1;
    while (rem >= (M1 - 1 - u)) { rem -= (M1 - 1 - u); ++u; }
    lut1[k1] = (unsigned short)(u | ((u + 1 + rem) << 8));
  }
}

// Uniform GEMM consumer: A fragments from the shared feature buffer,
// B fragments from global weights (L2-resident).
__device__ __forceinline__ void consume(const float* __restrict__ wseg, int kg0,
                                        int kc, const float* __restrict__ fbuf,
                                        int laneM, int laneH, int col,
                                        v8f& acc0, v8f& acc1) {
  for (int kb = 0; kb < kc; kb += 4) {
    const int kk = kb + 2 * laneH;
    v2f B = {wseg[(size_t)(kg0 + kk) * M0 + col],
             wseg[(size_t)(kg0 + kk + 1) * M0 + col]};
    v2f Aa = *(const v2f*)(fbuf + laneM * FBS + kk);
    v2f Ab = *(const v2f*)(fbuf + (16 + laneM) * FBS + kk);
    acc0 = wmma4(Aa, B, acc0);
    acc1 = wmma4(Ab, B, acc1);
  }
}

// ---------------------------------------------------------------------------
// Kernel 1: out0 (cols 0..127). Four fused GEMM segments, K total = 10336.
// Block = 256 threads (8 waves). Features are produced cooperatively into a
// chunked LDS buffer (scales folded), then all 8 waves consume them as WMMA
// A-fragments against their 16 columns of B.
// ---------------------------------------------------------------------------
__global__ __launch_bounds__(256) void out0_kernel(
    const float* __restrict__ x, const float* __restrict__ w1,
    const float* __restrict__ w2, const float* __restrict__ w4,
    const float* __restrict__ w5, const unsigned short* __restrict__ lut0,
    const unsigned short* __restrict__ lut1, float* __restrict__ out,
    float s1, float s2, float s4, float s5) {
  __shared__ float x0s[R2 * X0S];
  __shared__ float x1s[R2 * X1S];
  __shared__ float fbuf[R2 * FBS];
  const int tid = threadIdx.x;
  const int r0 = blockIdx.x * R2;

  // Stage x rows: split into x0 (128) and x1 (192) LDS tiles, float4 loads.
  for (int q = tid; q < R2 * (XW / 4); q += 256) {
    const int row = q / (XW / 4);
    const int c = (q % (XW / 4)) * 4;
    const float4 v = *(const float4*)(x + (size_t)(r0 + row) * XW + c);
    if (c < M0) *(float4*)(x0s + row * X0S + c) = v;
    else        *(float4*)(x1s + row * X1S + (c - M0)) = v;
  }

  const int wave = tid >> 5, lane = tid & 31;
  const int laneM = lane & 15, laneH = lane >> 4;
  const int col = wave * 16 + laneM;
  v8f acc0 = {}, acc1 = {};

  // --- segment 1: C1 * (x0[:,IU0]*x0[:,IV0]) @ w1, K = 8128, chunks of 128 ---
  for (int kb0 = 0; kb0 < NP0; kb0 += KC) {
    const int kc = (NP0 - kb0 < KC) ? (NP0 - kb0) : KC;  // 128 ... tail 64
    __syncthreads();
    for (int idx = tid; idx < R2 * kc; idx += 256) {
      const int m = idx / kc, j = idx - m * kc;
      const unsigned p = lut0[kb0 + j];
      const float* r = x0s + m * X0S;
      fbuf[m * FBS + j] = s1 * r[p & 255] * r[p >> 8];
    }
    __syncthreads();
    consume(w1, kb0, kc, fbuf, laneM, laneH, col, acc0, acc1);
  }
  // --- segment 2: C2 * (x0*x0) @ w2, K = 128 (one chunk) ---
  __syncthreads();
  for (int idx = tid; idx < R2 * M0; idx += 256) {
    const int m = idx >> 7, j = idx & 127;
    const float v = x0s[m * X0S + j];
    fbuf[m * FBS + j] = s2 * v * v;
  }
  __syncthreads();
  consume(w2, 0, M0, fbuf, laneM, laneH, col, acc0, acc1);
  // --- segment 3: C4/sqrt3 * s_pair @ w4, K = 2016, chunks of 96 ---
  for (int kb0 = 0; kb0 < NP1; kb0 += 96) {
    __syncthreads();
    for (int idx = tid; idx < R2 * 96; idx += 256) {
      const int m = idx / 96, j = idx - m * 96;
      const unsigned p = lut1[kb0 + j];
      const float* r = x1s + m * X1S;
      const int u = 3 * (p & 255), v = 3 * (p >> 8);
      fbuf[m * FBS + j] =
          s4 * (r[u] * r[v] + r[u + 1] * r[v + 1] + r[u + 2] * r[v + 2]);
    }
    __syncthreads();
    consume(w4, kb0, 96, fbuf, laneM, laneH, col, acc0, acc1);
  }
  // --- segment 4: C5/sqrt3 * row-norms @ w5, K = 64 (one chunk) ---
  __syncthreads();
  for (int idx = tid; idx < R2 * M1; idx += 256) {
    const int m = idx >> 6, j = idx & 63;
    const float* r = x1s + m * X1S + 3 * j;
    fbuf[m * FBS + j] = s5 * (r[0] * r[0] + r[1] * r[1] + r[2] * r[2]);
  }
  __syncthreads();
  consume(w5, 0, M1, fbuf, laneM, laneH, col, acc0, acc1);

  // D layout: VGPR g -> M = g + 8*laneH, N = laneM
#pragma unroll
  for (int g = 0; g < 8; ++g) {
    out[(size_t)(r0 + g + 8 * laneH) * OW + col] = acc0[g];
    out[(size_t)(r0 + 16 + g + 8 * laneH) * OW + col] = acc1[g];
  }
}

// ---------------------------------------------------------------------------
// Kernel 2: out1 (cols 128..319). t[n,v,w] = x0 @ w3[:,v,:] via WMMA (K=128),
// then out1[n,w,k] += t[n,v,w]*x1[n,v,k]. 8 waves = 4 w-tiles x 2 v-halves.
// A fragments (x0 rows) are v-invariant -> preloaded into 32 v2f registers.
// ---------------------------------------------------------------------------
__global__ __launch_bounds__(256) void out1_kernel(
    const float* __restrict__ x, const float* __restrict__ w3,
    float* __restrict__ out, float s3) {
  __shared__ float x0s[R3 * X0S];
  __shared__ float x1s[R3 * X1S];
  __shared__ float red[4 * 3 * 256];
  const int tid = threadIdx.x;
  const int r0 = blockIdx.x * R3;

  for (int q = tid; q < R3 * (XW / 4); q += 256) {
    const int row = q / (XW / 4);
    const int c = (q % (XW / 4)) * 4;
    const float4 v = *(const float4*)(x + (size_t)(r0 + row) * XW + c);
    if (c < M0) *(float4*)(x0s + row * X0S + c) = v;
    else        *(float4*)(x1s + row * X1S + (c - M0)) = v;
  }
  __syncthreads();

  const int wave = tid >> 5, lane = tid & 31;
  const int laneM = lane & 15, laneH = lane >> 4;
  const int wtile = wave & 3;   // which 16-wide w tile
  const int vhalf = wave >> 2;  // which half of v range
  const int wcol = wtile * 16 + laneM;
  const float* Ar = x0s + laneM * X0S;

  // Preload the 32 K-step A fragments (invariant over the whole v loop).
  v2f Af[32];
#pragma unroll
  for (int s = 0; s < 32; ++s)
    Af[s] = *(const v2f*)(Ar + 4 * s + 2 * laneH);

  v8f a0 = {}, a1 = {}, a2 = {};
  for (int v = vhalf * 32; v < vhalf * 32 + 32; ++v) {
    v8f t = {};
    const float* Bb = w3 + (size_t)v * 64 + wcol;  // w3[u][v][w], u-stride 4096
#pragma unroll
    for (int s = 0; s < 32; ++s) {
      const int kk = 4 * s + 2 * laneH;
      v2f B = {Bb[(size_t)kk * 4096], Bb[(size_t)(kk + 1) * 4096]};
      t = wmma4(Af[s], B, t);
    }
#pragma unroll
    for (int g = 0; g < 8; ++g) {
      const float* xr = x1s + (g + 8 * laneH) * X1S + 3 * v;
      a0[g] += t[g] * xr[0];
      a1[g] += t[g] * xr[1];
      a2[g] += t[g] * xr[2];
    }
  }
  __syncthreads();
  if (vhalf == 1) {
#pragma unroll
    for (int g = 0; g < 8; ++g) {
      red[(wtile * 3 + 0) * 256 + g * 32 + lane] = a0[g];
      red[(wtile * 3 + 1) * 256 + g * 32 + lane] = a1[g];
      red[(wtile * 3 + 2) * 256 + g * 32 + lane] = a2[g];
    }
  }
  __syncthreads();
  if (vhalf == 0) {
#pragma unroll
    for (int g = 0; g < 8; ++g) {
      const float o0 = a0[g] + red[(wtile * 3 + 0) * 256 + g * 32 + lane];
      const float o1 = a1[g] + red[(wtile * 3 + 1) * 256 + g * 32 + lane];
      const float o2 = a2[g] + red[(wtile * 3 + 2) * 256 + g * 32 + lane];
      const size_t base = (size_t)(r0 + g + 8 * laneH) * OW + M0 + wcol * 3;
      out[base + 0] = s3 * o0;
      out[base + 1] = s3 * o1;
      out[base + 2] = s3 * o2;
    }
  }
}

// ---------------------------------------------------------------------------
extern "C" void kernel_launch(void* const* d_in, const int* in_sizes, int n_in,
                              void* d_out, int out_size, void* d_ws,
                              size_t ws_size, hipStream_t stream) {
  const float* x  = (const float*)d_in[0];
  const float* w1 = (const float*)d_in[1];
  const float* w2 = (const float*)d_in[2];
  const float* w3 = (const float*)d_in[3];
  const float* w4 = (const float*)d_in[4];
  const float* w5 = (const float*)d_in[5];
  float* out = (float*)d_out;
  const int n = in_sizes[0] / XW;  // 8192

  unsigned short* lut0 = (unsigned short*)d_ws;
  unsigned short* lut1 = lut0 + NP0;

  const float inv_sqrt3 = 0.57735026918962576f;
  const float Xo = (float)(NP0 + M0 + NP1 + M1);  // 10336
  const float s1 = sqrtf(1.0f / Xo);
  const float s2 = sqrtf(1.0f / (3.0f * Xo));
  const float s4 = s1 * inv_sqrt3;
  const float s5 = sqrtf(1.0f / (5.0f * Xo)) * inv_sqrt3;
  const float s3 = sqrtf(3.0f / (float)(M0 * M1)) * inv_sqrt3;  // = 1/sqrt(8192)

  build_luts<<<(NP0 + NP1 + 255) / 256, 256, 0, stream>>>(lut0, lut1);
  out0_kernel<<<n / R2, 256, 0, stream>>>(x, w1, w2, w4, w5, lut0, lut1, out,
                                          s1, s2, s4, s5);
  out1_kernel<<<n / R3, 256, 0, stream>>>(x, w3, out, s3);
}